// RewardNet_89395449299690
// MI455X (gfx1250) — compile-verified
//
#include <hip/hip_runtime.h>
#include <math.h>

// ---------------------------------------------------------------------------
// CDNA5 (gfx1250) wave32 WMMA fp32 implementation of the RewardNet forward.
// All heavy GEMMs + attention use V_WMMA_F32_16X16X4_F32 (full fp32 precision,
// matches the fp32 reference numerics while using the matrix pipe).
// ---------------------------------------------------------------------------

typedef __attribute__((ext_vector_type(2))) float v2f;
typedef __attribute__((ext_vector_type(8))) float v8f;

__device__ inline v8f v8zero() {
  v8f z = {0.f,0.f,0.f,0.f,0.f,0.f,0.f,0.f};
  return z;
}

// 8-arg form: (neg_a, A, neg_b, B, c_mod, C, reuse_a, reuse_b)
#define WMMA_F32X4(a,b,c) \
  __builtin_amdgcn_wmma_f32_16x16x4_f32(false,(a),false,(b),(short)0,(c),false,false)

// ---------------------------------------------------------------------------
// block reduction helpers (blockDim must be a power of two)
// ---------------------------------------------------------------------------
__device__ inline float block_sum(float v, float* red) {
  const int tid = threadIdx.x;
  red[tid] = v;
  __syncthreads();
  for (int s = blockDim.x >> 1; s > 0; s >>= 1) {
    if (tid < s) red[tid] += red[tid + s];
    __syncthreads();
  }
  float r = red[0];
  __syncthreads();
  return r;
}

__device__ inline float block_fmax(float v, float* red) {
  const int tid = threadIdx.x;
  red[tid] = v;
  __syncthreads();
  for (int s = blockDim.x >> 1; s > 0; s >>= 1) {
    if (tid < s) red[tid] = fmaxf(red[tid], red[tid + s]);
    __syncthreads();
  }
  float r = red[0];
  __syncthreads();
  return r;
}

// ---------------------------------------------------------------------------
// WMMA GEMM:  C[M,N] = op(A)[M,K] @ B[K,N] (+bias)(+bias2)(relu?)
// MODE 0: A[m*lda + k]
// MODE 1: A[m*lda + k] * scale[m]          (row-normalized se_weights)
// MODE 2: A[k*lda + m] * scale[m]          (col-normalized se_weights^T)
// One wave computes a 16x64 tile (A fragment reused over 4 N-subtiles).
// Requires: M%16==0, N%64==0, K%4==0. grid = (N/64, M/16), block = 32.
// ---------------------------------------------------------------------------
template<int MODE, bool RELU>
__global__ __launch_bounds__(32) void wmma_gemm_kernel(
    const float* __restrict__ A, int lda, const float* __restrict__ scale,
    const float* __restrict__ B,
    const float* __restrict__ bias, const float* __restrict__ bias2,
    float* __restrict__ C, int M, int N, int K) {
  const int lane = threadIdx.x & 31;
  const int hf   = lane >> 4;       // 0: k pair {0,1}, 1: k pair {2,3}
  const int r    = lane & 15;
  const int m0   = blockIdx.y * 16;
  const int n0   = blockIdx.x * 64;
  const int mrow = m0 + r;

  float smul = 1.0f;
  if (MODE != 0) smul = scale[mrow];

  v8f acc[4];
#pragma unroll
  for (int t = 0; t < 4; ++t) acc[t] = v8zero();

  for (int k0 = 0; k0 < K; k0 += 4) {
    const int ka = k0 + 2 * hf;
    v2f a;
    if (MODE == 2) {
      a.x = A[(size_t)ka * lda + mrow] * smul;
      a.y = A[(size_t)(ka + 1) * lda + mrow] * smul;
    } else {
      const float* ar = A + (size_t)mrow * lda + ka;
      a.x = ar[0];
      a.y = ar[1];
      if (MODE == 1) { a.x *= smul; a.y *= smul; }
    }
#pragma unroll
    for (int t = 0; t < 4; ++t) {
      const float* bp = B + (size_t)ka * N + n0 + t * 16 + r;
      v2f b;
      b.x = bp[0];
      b.y = bp[N];
      acc[t] = WMMA_F32X4(a, b, acc[t]);
    }
  }

#pragma unroll
  for (int t = 0; t < 4; ++t) {
    const int col = n0 + t * 16 + r;
    float bv = 0.f;
    if (bias)  bv  = bias[col];
    if (bias2) bv += bias2[col];
#pragma unroll
    for (int v = 0; v < 8; ++v) {
      const int mo = m0 + v + 8 * hf;
      float val = acc[t][v] + bv;
      if (RELU) val = fmaxf(val, 0.f);
      C[(size_t)mo * N + col] = val;
    }
  }
}

// ---------------------------------------------------------------------------
// Fused multi-head attention (flash-style online softmax).
// qkv: [T,768] rows = [q(256) | k(256) | v(256)], H=4, hd=64.
// One wave handles (head, 16-query tile); loops over key tiles of 16.
// out: [T,256]. grid = (T/16, 4), block = 32.
// ---------------------------------------------------------------------------
__global__ __launch_bounds__(32) void attn_kernel(
    const float* __restrict__ qkv, const float* __restrict__ mask,
    float* __restrict__ out, int T) {
  const int h    = blockIdx.y;
  const int qt   = blockIdx.x * 16;
  const int lane = threadIdx.x & 31;
  const int hf   = lane >> 4;
  const int r    = lane & 15;
  const float scl = 0.125f;  // 1/sqrt(64)

  __shared__ float s_p[16][17];
  __shared__ float s_m[16], s_l[16], s_f[16];

  if (lane < 16) { s_m[lane] = -1e30f; s_l[lane] = 0.f; }

  const float* Q  = qkv + h * 64;
  const float* Kp = qkv + 256 + h * 64;
  const float* Vp = qkv + 512 + h * 64;

  // preload q fragments (16 K-chunks of 4 covering hd=64)
  v2f qa[16];
  {
    const float* qr = Q + (size_t)(qt + r) * 768;
#pragma unroll
    for (int kc = 0; kc < 16; ++kc) {
      qa[kc].x = qr[kc * 4 + 2 * hf];
      qa[kc].y = qr[kc * 4 + 2 * hf + 1];
    }
  }

  v8f o[4];
#pragma unroll
  for (int t = 0; t < 4; ++t) o[t] = v8zero();
  __syncthreads();

  for (int kt = 0; kt < T; kt += 16) {
    // scores s = q @ k^T  (16x16)
    v8f s = v8zero();
    const float* kr = Kp + (size_t)(kt + r) * 768;
#pragma unroll
    for (int kc = 0; kc < 16; ++kc) {
      v2f b;
      b.x = kr[kc * 4 + 2 * hf];
      b.y = kr[kc * 4 + 2 * hf + 1];
      s = WMMA_F32X4(qa[kc], b, s);
    }
#pragma unroll
    for (int v = 0; v < 8; ++v) {
      const int qrow = v + 8 * hf;
      s_p[qrow][r] = s[v] * scl + mask[(size_t)(qt + qrow) * T + kt + r];
    }
    __syncthreads();

    // online softmax update, one thread per query row
    if (lane < 16) {
      const float mold = s_m[lane];
      float mt = mold;
#pragma unroll
      for (int c = 0; c < 16; ++c) mt = fmaxf(mt, s_p[lane][c]);
      const float f = expf(mold - mt);
      float l = s_l[lane] * f;
#pragma unroll
      for (int c = 0; c < 16; ++c) {
        const float p = expf(s_p[lane][c] - mt);
        s_p[lane][c] = p;
        l += p;
      }
      s_m[lane] = mt; s_l[lane] = l; s_f[lane] = f;
    }
    __syncthreads();

    // rescale accumulators by exp(m_old - m_new)
#pragma unroll
    for (int t = 0; t < 4; ++t)
#pragma unroll
      for (int v = 0; v < 8; ++v)
        o[t][v] *= s_f[v + 8 * hf];

    // o += p @ v_tile  (16x16 @ 16x64)
#pragma unroll
    for (int kc = 0; kc < 4; ++kc) {
      const int ka = kc * 4 + 2 * hf;
      v2f a;
      a.x = s_p[r][ka];
      a.y = s_p[r][ka + 1];
      const float* vr0 = Vp + (size_t)(kt + ka) * 768;
      const float* vr1 = Vp + (size_t)(kt + ka + 1) * 768;
#pragma unroll
      for (int t = 0; t < 4; ++t) {
        v2f b;
        b.x = vr0[t * 16 + r];
        b.y = vr1[t * 16 + r];
        o[t] = WMMA_F32X4(a, b, o[t]);
      }
    }
    __syncthreads();
  }

#pragma unroll
  for (int t = 0; t < 4; ++t)
#pragma unroll
    for (int v = 0; v < 8; ++v) {
      const int qrow = v + 8 * hf;
      out[(size_t)(qt + qrow) * 256 + h * 64 + t * 16 + r] = o[t][v] / s_l[qrow];
    }
}

// ---------------------------------------------------------------------------
// LayerNorm over D (<=1024), optional residual add. In-place safe (row cached).
// grid = rows, block = 256.
// ---------------------------------------------------------------------------
template<bool ADD>
__global__ __launch_bounds__(256) void ln_row_kernel(
    const float* __restrict__ x, const float* __restrict__ y,
    const float* __restrict__ g, const float* __restrict__ b,
    float* __restrict__ out, int D) {
  __shared__ float cache[1024];
  __shared__ float red[256];
  const int row = blockIdx.x, tid = threadIdx.x;
  float sum = 0.f;
  for (int i = tid; i < D; i += 256) {
    float v = x[(size_t)row * D + i];
    if (ADD) v += y[(size_t)row * D + i];
    cache[i] = v;
    sum += v;
  }
  const float mean = block_sum(sum, red) / (float)D;
  float vs = 0.f;
  for (int i = tid; i < D; i += 256) {
    const float d = cache[i] - mean;
    vs += d * d;
  }
  const float var = block_sum(vs, red) / (float)D;
  const float rstd = rsqrtf(var + 1e-5f);
  for (int i = tid; i < D; i += 256)
    out[(size_t)row * D + i] = (cache[i] - mean) * rstd * g[i] + b[i];
}

// LN over concat(a,c) (256+256=512). grid = rows, block = 256. out: [rows,512]
__global__ __launch_bounds__(256) void ln_concat_kernel(
    const float* __restrict__ a, const float* __restrict__ c,
    const float* __restrict__ g, const float* __restrict__ b,
    float* __restrict__ out) {
  __shared__ float cache[512];
  __shared__ float red[256];
  const int row = blockIdx.x, tid = threadIdx.x;
  const float v0 = a[(size_t)row * 256 + tid];
  const float v1 = c[(size_t)row * 256 + tid];
  cache[tid] = v0;
  cache[256 + tid] = v1;
  const float mean = block_sum(v0 + v1, red) / 512.f;
  const float d0 = v0 - mean, d1 = v1 - mean;
  const float var = block_sum(d0 * d0 + d1 * d1, red) / 512.f;
  const float rstd = rsqrtf(var + 1e-5f);
  out[(size_t)row * 512 + tid]       = d0 * rstd * g[tid] + b[tid];
  out[(size_t)row * 512 + 256 + tid] = d1 * rstd * g[256 + tid] + b[256 + tid];
}

// ---------------------------------------------------------------------------
// MLP encoder: LN(din) -> relu(W1) -> relu(W2) -> LN(256). One row per block.
// ---------------------------------------------------------------------------
__global__ __launch_bounds__(256) void mlp_enc_kernel(
    const float* __restrict__ in, int din,
    const float* __restrict__ lg, const float* __restrict__ lb,
    const float* __restrict__ W1, const float* __restrict__ b1,
    const float* __restrict__ W2, const float* __restrict__ b2,
    const float* __restrict__ og, const float* __restrict__ ob,
    float* __restrict__ out) {
  const int row = blockIdx.x, tid = threadIdx.x;
  __shared__ float xn[16];
  __shared__ float h1[256];
  __shared__ float red[256];
  if (tid == 0) {
    float m = 0.f;
    for (int i = 0; i < din; ++i) m += in[(size_t)row * din + i];
    m /= (float)din;
    float v = 0.f;
    for (int i = 0; i < din; ++i) {
      const float d = in[(size_t)row * din + i] - m;
      v += d * d;
    }
    const float rstd = rsqrtf(v / (float)din + 1e-5f);
    for (int i = 0; i < din; ++i)
      xn[i] = (in[(size_t)row * din + i] - m) * rstd * lg[i] + lb[i];
  }
  __syncthreads();
  float s = b1[tid];
  for (int i = 0; i < din; ++i) s += xn[i] * W1[i * 256 + tid];
  h1[tid] = fmaxf(s, 0.f);
  __syncthreads();
  s = b2[tid];
  for (int i = 0; i < 256; ++i) s += h1[i] * W2[i * 256 + tid];
  const float h2 = fmaxf(s, 0.f);
  const float mean = block_sum(h2, red) / 256.f;
  const float d = h2 - mean;
  const float var = block_sum(d * d, red) / 256.f;
  out[(size_t)row * 256 + tid] = d * rsqrtf(var + 1e-5f) * og[tid] + ob[tid];
}

// ---------------------------------------------------------------------------
// se_weights row / column normalizers
// ---------------------------------------------------------------------------
__global__ __launch_bounds__(256) void rowinv_kernel(
    const float* __restrict__ se, float* __restrict__ rinv, int E) {
  __shared__ float red[256];
  const int row = blockIdx.x, tid = threadIdx.x;
  float s = 0.f;
  for (int i = tid; i < E; i += 256) s += se[(size_t)row * E + i];
  s = block_sum(s, red);
  if (tid == 0) rinv[row] = 1.f / fmaxf(s, 1e-8f);
}

__global__ __launch_bounds__(256) void colinv_kernel(
    const float* __restrict__ se, float* __restrict__ cinv, int S, int E) {
  const int e = blockIdx.x * 256 + threadIdx.x;
  if (e >= E) return;
  float s = 0.f;
  for (int i = 0; i < S; ++i) s += se[(size_t)i * E + e];
  cinv[e] = 1.f / fmaxf(s, 1e-8f);
}

// ---------------------------------------------------------------------------
// Attention pooling: per-row score = tanh(LN(x)@W1+b1)@W2+b2.  block=128.
// ---------------------------------------------------------------------------
__global__ __launch_bounds__(128) void pool_score_kernel(
    const float* __restrict__ x,
    const float* __restrict__ lg, const float* __restrict__ lb,
    const float* __restrict__ W1, const float* __restrict__ b1,
    const float* __restrict__ W2, const float* __restrict__ b2,
    float* __restrict__ scores) {
  const int row = blockIdx.x, tid = threadIdx.x;
  __shared__ float xr[256];
  __shared__ float red[128];
  const float v0 = x[(size_t)row * 256 + tid];
  const float v1 = x[(size_t)row * 256 + 128 + tid];
  const float mean = block_sum(v0 + v1, red) / 256.f;
  const float d0 = v0 - mean, d1 = v1 - mean;
  const float var = block_sum(d0 * d0 + d1 * d1, red) / 256.f;
  const float rstd = rsqrtf(var + 1e-5f);
  xr[tid]       = d0 * rstd * lg[tid] + lb[tid];
  xr[128 + tid] = d1 * rstd * lg[128 + tid] + lb[128 + tid];
  __syncthreads();
  float h = b1[tid];
  for (int i = 0; i < 256; ++i) h += xr[i] * W1[i * 128 + tid];
  h = tanhf(h);
  const float c = block_sum(h * W2[tid], red);
  if (tid == 0) scores[row] = c + b2[0];
}

// softmax over T scores, then [attn_pool | mean | max] -> out[768]. block=256.
__global__ __launch_bounds__(256) void pool_finalize_kernel(
    const float* __restrict__ x, const float* __restrict__ scores,
    int T, float* __restrict__ out) {
  __shared__ float red[256];
  const int tid = threadIdx.x;
  float mx = -1e30f;
  for (int t = tid; t < T; t += 256) mx = fmaxf(mx, scores[t]);
  const float smax = block_fmax(mx, red);
  float se = 0.f;
  for (int t = tid; t < T; t += 256) se += expf(scores[t] - smax);
  const float sinv = 1.f / block_sum(se, red);
  float ap = 0.f, mn = 0.f, mv = -1e30f;
  for (int t = 0; t < T; ++t) {
    const float v = x[(size_t)t * 256 + tid];
    const float a = expf(scores[t] - smax) * sinv;
    ap += a * v;
    mn += v;
    mv = fmaxf(mv, v);
  }
  out[tid]       = ap;
  out[256 + tid] = mn / (float)T;
  out[512 + tid] = mv;
}

// out[n] = b[n] + sum_k v[k]*W[k*N+n]. block=256, grid=N/256.
__global__ __launch_bounds__(256) void vecmat_kernel(
    const float* __restrict__ v, const float* __restrict__ W,
    const float* __restrict__ b, float* __restrict__ out, int K, int N) {
  const int n = blockIdx.x * 256 + threadIdx.x;
  if (n >= N) return;
  float s = b[n];
  for (int k = 0; k < K; ++k) s += v[k] * W[(size_t)k * N + n];
  out[n] = s;
}

// feat = [s, e, |s-e|, s*e]  (4x256). grid=1, block=256.
__global__ __launch_bounds__(256) void feat_kernel(
    const float* __restrict__ s_emb, const float* __restrict__ e_emb,
    float* __restrict__ feat) {
  const int i = threadIdx.x;
  const float a = s_emb[i], b = e_emb[i];
  feat[i] = a;
  feat[256 + i] = b;
  feat[512 + i] = fabsf(a - b);
  feat[768 + i] = a * b;
}

// out head: LN(1024)->relu(1024x512)->relu(512x128)->128x1. grid=1, block=512.
__global__ __launch_bounds__(512) void head_kernel(
    const float* __restrict__ feat,
    const float* __restrict__ lg, const float* __restrict__ lb,
    const float* __restrict__ W1, const float* __restrict__ b1,
    const float* __restrict__ W2, const float* __restrict__ b2,
    const float* __restrict__ W3, const float* __restrict__ b3,
    float* __restrict__ out) {
  const int tid = threadIdx.x;
  __shared__ float lnf[1024];
  __shared__ float h1[512];
  __shared__ float h2[128];
  __shared__ float red[512];
  float s = 0.f;
  for (int i = tid; i < 1024; i += 512) {
    const float v = feat[i];
    lnf[i] = v;
    s += v;
  }
  const float mean = block_sum(s, red) / 1024.f;
  float vs = 0.f;
  for (int i = tid; i < 1024; i += 512) {
    const float d = lnf[i] - mean;
    vs += d * d;
  }
  const float var = block_sum(vs, red) / 1024.f;
  const float rstd = rsqrtf(var + 1e-5f);
  for (int i = tid; i < 1024; i += 512)
    lnf[i] = (lnf[i] - mean) * rstd * lg[i] + lb[i];
  __syncthreads();
  float a = b1[tid];
  for (int i = 0; i < 1024; ++i) a += lnf[i] * W1[i * 512 + tid];
  h1[tid] = fmaxf(a, 0.f);
  __syncthreads();
  if (tid < 128) {
    float a2 = b2[tid];
    for (int i = 0; i < 512; ++i) a2 += h1[i] * W2[i * 128 + tid];
    h2[tid] = fmaxf(a2, 0.f);
  }
  __syncthreads();
  const float c = (tid < 128) ? h2[tid] * W3[tid] : 0.f;
  const float r = block_sum(c, red);
  if (tid == 0) out[0] = r + b3[0];
}

// ---------------------------------------------------------------------------
// host-side parameter structs + encoder layer driver
// ---------------------------------------------------------------------------
struct MlpEncP { const float *lg,*lb,*W1,*b1,*W2,*b2,*og,*ob; };
struct LayerP  { const float *Wqkv,*bqkv,*Wo,*bo,*l1g,*l1b,*W1,*b1,*W2,*b2,*l2g,*l2b; };
struct FuseP   { const float *lg,*lb,*W,*b; };
struct PoolP   { const float *lg,*lb,*W1,*b1,*W2,*b2; };
struct HeadP   { const float *lg,*lb,*W1,*b1,*W2,*b2,*W3,*b3; };

static void run_encoder_layer(float* x, const float* mask, int T, const LayerP& L,
                              float* qkv, float* attn_o, float* tmp, float* ffn,
                              hipStream_t stream) {
  // qkv = x @ Wqkv + bqkv           (T,256)@(256,768)
  wmma_gemm_kernel<0,false><<<dim3(768/64, T/16), 32, 0, stream>>>(
      x, 256, nullptr, L.Wqkv, L.bqkv, nullptr, qkv, T, 768, 256);
  // fused MHA
  attn_kernel<<<dim3(T/16, 4), 32, 0, stream>>>(qkv, mask, attn_o, T);
  // tmp = attn_o @ Wo + bo
  wmma_gemm_kernel<0,false><<<dim3(256/64, T/16), 32, 0, stream>>>(
      attn_o, 256, nullptr, L.Wo, L.bo, nullptr, tmp, T, 256, 256);
  // x = LN(x + tmp)
  ln_row_kernel<true><<<T, 256, 0, stream>>>(x, tmp, L.l1g, L.l1b, x, 256);
  // ffn = relu(x @ W1 + b1)         (T,256)@(256,1024)
  wmma_gemm_kernel<0,true><<<dim3(1024/64, T/16), 32, 0, stream>>>(
      x, 256, nullptr, L.W1, L.b1, nullptr, ffn, T, 1024, 256);
  // tmp = ffn @ W2 + b2             (T,1024)@(1024,256)
  wmma_gemm_kernel<0,false><<<dim3(256/64, T/16), 32, 0, stream>>>(
      ffn, 1024, nullptr, L.W2, L.b2, nullptr, tmp, T, 256, 1024);
  // x = LN(x + tmp)
  ln_row_kernel<true><<<T, 256, 0, stream>>>(x, tmp, L.l2g, L.l2b, x, 256);
}

// ---------------------------------------------------------------------------
extern "C" void kernel_launch(void* const* d_in, const int* in_sizes, int n_in,
                              void* d_out, int out_size, void* d_ws, size_t ws_size,
                              hipStream_t stream) {
  (void)in_sizes; (void)n_in; (void)out_size; (void)ws_size;
  const int S = 1024, E = 2048, TT = 3072;  // S, E, S+E

  const float* sub_feat    = (const float*)d_in[0];  // (S,5)
  const float* edge_feat   = (const float*)d_in[1];  // (E,10)
  const float* se          = (const float*)d_in[2];  // (S,E)
  const float* edge_mask   = (const float*)d_in[3];  // (E,E)
  const float* sparse_mask = (const float*)d_in[4];  // (TT,TT)

  // ---- params: one leaf per pointer, depth-first insertion order ----
  int idx = 5;
  auto nx = [&]() { return (const float*)d_in[idx++]; };

  MlpEncP subE; subE.lg=nx(); subE.lb=nx(); subE.W1=nx(); subE.b1=nx();
                subE.W2=nx(); subE.b2=nx(); subE.og=nx(); subE.ob=nx();
  MlpEncP edgE; edgE.lg=nx(); edgE.lb=nx(); edgE.W1=nx(); edgE.b1=nx();
                edgE.W2=nx(); edgE.b2=nx(); edgE.og=nx(); edgE.ob=nx();

  const float *gpre_g=nx(), *gpre_b=nx(), *gpost_g=nx(), *gpost_b=nx();
  LayerP gl[2];
  for (int i = 0; i < 2; ++i) {
    LayerP& L = gl[i];
    L.Wqkv=nx(); L.bqkv=nx(); L.Wo=nx(); L.bo=nx(); L.l1g=nx(); L.l1b=nx();
    L.W1=nx(); L.b1=nx(); L.W2=nx(); L.b2=nx(); L.l2g=nx(); L.l2b=nx();
  }
  FuseP fsub; fsub.lg=nx(); fsub.lb=nx(); fsub.W=nx(); fsub.b=nx();
  FuseP fedg; fedg.lg=nx(); fedg.lb=nx(); fedg.W=nx(); fedg.b=nx();

  const float *Win=nx(), *bin=nx();
  const float *spre_g=nx(), *spre_b=nx(), *spost_g=nx(), *spost_b=nx();
  LayerP sl[2];
  for (int i = 0; i < 2; ++i) {
    LayerP& L = sl[i];
    L.Wqkv=nx(); L.bqkv=nx(); L.Wo=nx(); L.bo=nx(); L.l1g=nx(); L.l1b=nx();
    L.W1=nx(); L.b1=nx(); L.W2=nx(); L.b2=nx(); L.l2g=nx(); L.l2b=nx();
  }
  PoolP psub; psub.lg=nx(); psub.lb=nx(); psub.W1=nx(); psub.b1=nx(); psub.W2=nx(); psub.b2=nx();
  PoolP pedg; pedg.lg=nx(); pedg.lb=nx(); pedg.W1=nx(); pedg.b1=nx(); pedg.W2=nx(); pedg.b2=nx();
  const float *projSW=nx(), *projSb=nx(), *projEW=nx(), *projEb=nx();
  HeadP hp; hp.lg=nx(); hp.lb=nx(); hp.W1=nx(); hp.b1=nx();
            hp.W2=nx(); hp.b2=nx(); hp.W3=nx(); hp.b3=nx();

  // ---- workspace layout (floats), ~46 MB total ----
  float* ws = (float*)d_ws;
  size_t off = 0;
  auto alloc = [&](size_t n) { float* p = ws + off; off += n; return p; };
  float* sub_repr  = alloc((size_t)S * 256);
  float* edge_repr = alloc((size_t)E * 256);
  float* x         = alloc((size_t)TT * 256);
  float* qkv       = alloc((size_t)TT * 768);
  float* attn_o    = alloc((size_t)TT * 256);
  float* tmp       = alloc((size_t)TT * 256);
  float* ffn       = alloc((size_t)TT * 1024);
  float* rinv      = alloc(S);
  float* cinv      = alloc(E);
  float* se_sub    = alloc((size_t)S * 256);
  float* se_edge   = alloc((size_t)E * 256);
  float* zcat      = alloc((size_t)E * 512);
  float* sub_f     = alloc((size_t)S * 256);
  float* edge_f    = alloc((size_t)E * 256);
  float* scoresS   = alloc(S);
  float* scoresE   = alloc(E);
  float* poolS     = alloc(768);
  float* poolE     = alloc(768);
  float* sub_emb   = alloc(256);
  float* edge_emb  = alloc(256);
  float* feat      = alloc(1024);

  // ---- 1. MLP encoders ----
  mlp_enc_kernel<<<S, 256, 0, stream>>>(sub_feat, 5,
      subE.lg, subE.lb, subE.W1, subE.b1, subE.W2, subE.b2, subE.og, subE.ob, sub_repr);
  mlp_enc_kernel<<<E, 256, 0, stream>>>(edge_feat, 10,
      edgE.lg, edgE.lb, edgE.W1, edgE.b1, edgE.W2, edgE.b2, edgE.og, edgE.ob, edge_repr);

  // ---- 2. Graph transformer over edge tokens (T = E) ----
  ln_row_kernel<false><<<E, 256, 0, stream>>>(edge_repr, nullptr, gpre_g, gpre_b, x, 256);
  for (int i = 0; i < 2; ++i)
    run_encoder_layer(x, edge_mask, E, gl[i], qkv, attn_o, tmp, ffn, stream);
  ln_row_kernel<false><<<E, 256, 0, stream>>>(x, nullptr, gpost_g, gpost_b, edge_repr, 256);

  // ---- 3. se_weights normalized tokens ----
  rowinv_kernel<<<S, 256, 0, stream>>>(se, rinv, E);
  colinv_kernel<<<(E + 255) / 256, 256, 0, stream>>>(se, cinv, S, E);
  // se_sub = rownorm(se) @ edge_repr : (S,E)@(E,256)
  wmma_gemm_kernel<1,false><<<dim3(256/64, S/16), 32, 0, stream>>>(
      se, E, rinv, edge_repr, nullptr, nullptr, se_sub, S, 256, E);
  // se_edge = colnorm(se)^T @ sub_repr : (E,S)@(S,256)
  wmma_gemm_kernel<2,false><<<dim3(256/64, E/16), 32, 0, stream>>>(
      se, E, cinv, sub_repr, nullptr, nullptr, se_edge, E, 256, S);

  // ---- 4. fuse ----
  ln_concat_kernel<<<S, 256, 0, stream>>>(sub_repr, se_sub, fsub.lg, fsub.lb, zcat);
  wmma_gemm_kernel<0,true><<<dim3(256/64, S/16), 32, 0, stream>>>(
      zcat, 512, nullptr, fsub.W, fsub.b, nullptr, sub_f, S, 256, 512);
  ln_concat_kernel<<<E, 256, 0, stream>>>(edge_repr, se_edge, fedg.lg, fedg.lb, zcat);
  wmma_gemm_kernel<0,true><<<dim3(256/64, E/16), 32, 0, stream>>>(
      zcat, 512, nullptr, fedg.W, fedg.b, nullptr, edge_f, E, 256, 512);

  // ---- 5. tokens @ Win + bin; indicator one-hot folded into second bias ----
  // edge rows: ind=(0,1) -> + Win[257,:]; sub rows: ind=(1,0) -> + Win[256,:]
  wmma_gemm_kernel<0,false><<<dim3(256/64, E/16), 32, 0, stream>>>(
      edge_f, 256, nullptr, Win, bin, Win + (size_t)257 * 256, x, E, 256, 256);
  wmma_gemm_kernel<0,false><<<dim3(256/64, S/16), 32, 0, stream>>>(
      sub_f, 256, nullptr, Win, bin, Win + (size_t)256 * 256,
      x + (size_t)E * 256, S, 256, 256);

  // ---- 6. sparse transformer over all TT tokens ----
  ln_row_kernel<false><<<TT, 256, 0, stream>>>(x, nullptr, spre_g, spre_b, x, 256);
  for (int i = 0; i < 2; ++i)
    run_encoder_layer(x, sparse_mask, TT, sl[i], qkv, attn_o, tmp, ffn, stream);
  ln_row_kernel<false><<<TT, 256, 0, stream>>>(x, nullptr, spost_g, spost_b, x, 256);

  const float* edge_upd = x;                      // rows [0,E)
  const float* sub_upd  = x + (size_t)E * 256;    // rows [E,E+S)

  // ---- 7. pooling + projections ----
  pool_score_kernel<<<S, 128, 0, stream>>>(sub_upd,
      psub.lg, psub.lb, psub.W1, psub.b1, psub.W2, psub.b2, scoresS);
  pool_finalize_kernel<<<1, 256, 0, stream>>>(sub_upd, scoresS, S, poolS);
  pool_score_kernel<<<E, 128, 0, stream>>>(edge_upd,
      pedg.lg, pedg.lb, pedg.W1, pedg.b1, pedg.W2, pedg.b2, scoresE);
  pool_finalize_kernel<<<1, 256, 0, stream>>>(edge_upd, scoresE, E, poolE);
  vecmat_kernel<<<1, 256, 0, stream>>>(poolS, projSW, projSb, sub_emb, 768, 256);
  vecmat_kernel<<<1, 256, 0, stream>>>(poolE, projEW, projEb, edge_emb, 768, 256);

  // ---- 8. feature vector + output head ----
  feat_kernel<<<1, 256, 0, stream>>>(sub_emb, edge_emb, feat);
  head_kernel<<<1, 512, 0, stream>>>(feat,
      hp.lg, hp.lb, hp.W1, hp.b1, hp.W2, hp.b2, hp.W3, hp.b3, (float*)d_out);
}